// PhysioNeuron_3934190043678
// MI455X (gfx1250) — compile-verified
//
#include <hip/hip_runtime.h>
#include <hip/hip_bf16.h>
#include <stdint.h>

typedef float v2f __attribute__((ext_vector_type(2)));
typedef float v8f __attribute__((ext_vector_type(8)));
typedef int   v4i_vs __attribute__((vector_size(16)));  // matches builtin's b128 pointee type

#define B_ROWS 32768
#define DIM 1024
#define LDT 36   // LDS row stride (dwords): b128-store aligned, conflict-free b64 frag loads
#define LDT2 38  // LDS row stride for transposed staging (b32 writes), conflict-free

#if defined(__HIP_DEVICE_COMPILE__) &&                              \
    __has_builtin(__builtin_amdgcn_global_load_async_to_lds_b128) && \
    __has_builtin(__builtin_amdgcn_global_load_async_to_lds_b32) &&  \
    __has_builtin(__builtin_amdgcn_s_wait_asynccnt)
#define USE_ASYNC 1
#else
#define USE_ASYNC 0
#endif

// ---- global -> LDS copies (ASYNCcnt path on CDNA5, sync fallback otherwise) ----
__device__ __forceinline__ void cp_g2l_16(const float* g, float* l) {
#if USE_ASYNC
    __builtin_amdgcn_global_load_async_to_lds_b128(
        (__attribute__((address_space(1))) v4i_vs*)(uintptr_t)g,
        (__attribute__((address_space(3))) v4i_vs*)(uint32_t)(uintptr_t)l, 0, 0);
#else
    *(float4*)l = *(const float4*)g;
#endif
}
__device__ __forceinline__ void cp_g2l_4(const float* g, float* l) {
#if USE_ASYNC
    __builtin_amdgcn_global_load_async_to_lds_b32(
        (__attribute__((address_space(1))) int*)(uintptr_t)g,
        (__attribute__((address_space(3))) int*)(uint32_t)(uintptr_t)l, 0, 0);
#else
    *l = *g;
#endif
}

__device__ __forceinline__ v8f wmma4(v2f a, v2f b, v8f c) {
    // D = A(16x4,f32) * B(4x16,f32) + C(16x16,f32)
    return __builtin_amdgcn_wmma_f32_16x16x4_f32(false, a, false, b, (short)0, c, false, false);
}

__device__ __forceinline__ float wave_sum(float v) {
#pragma unroll
    for (int m = 16; m >= 1; m >>= 1) v += __shfl_xor(v, m, 32);
    return v;
}

__device__ __forceinline__ float sigmoidf(float z) {
    return 1.0f / (1.0f + __expf(-z));
}

// ---------------- K0: ||W_slow||_F^2 ----------------
__global__ __launch_bounds__(256) void wnorm_kernel(const float* __restrict__ W, float* __restrict__ scal) {
    const int lane = threadIdx.x & 31;
    int idx = blockIdx.x * 256 + threadIdx.x;
    const int stride = gridDim.x * 256;
    float s = 0.0f;
    for (int i = idx; i < DIM * DIM; i += stride) {
        float w = W[i];
        s += w * w;
    }
    s = wave_sum(s);
    if (lane == 0) atomicAdd(&scal[0], s);
}

// ---------------- K1: per-row variance of x (ddof=1) ----------------
__global__ __launch_bounds__(256) void rowstats_kernel(const float* __restrict__ X, float* __restrict__ stress) {
    const int tid = threadIdx.x, wave = tid >> 5, lane = tid & 31;
    const int row = blockIdx.x * 8 + wave;
    const float* xr = X + (size_t)row * DIM;
    float s = 0.0f, s2 = 0.0f;
#pragma unroll
    for (int j = 0; j < 8; j++) {
        float4 v = *(const float4*)(xr + j * 128 + lane * 4);
        s  += v.x + v.y + v.z + v.w;
        s2 += v.x * v.x + v.y * v.y + v.z * v.z + v.w * v.w;
    }
    s = wave_sum(s);
    s2 = wave_sum(s2);
    if (lane == 0)
        stress[row] = (s2 - s * s * (1.0f / DIM)) * (1.0f / (DIM - 1));
}

// ---------------- K2/K6: O = x @ W^T (WMMA f32, async double-buffered LDS) ----------------
// Block tile 64(M) x 128(N), K-block 32. 8 waves, wave = (m 4) x (n 2), each a 16x64 strip.
// FUSE=false: O = x@W^T.  FUSE=true: O += gate[row] * (x@W^T) (combined, in place over slow).
template <bool FUSE>
__global__ __launch_bounds__(256) void gemm_xw_kernel(const float* __restrict__ X,
                                                      const float* __restrict__ W,
                                                      const float* __restrict__ gate,
                                                      float* __restrict__ O) {
    __shared__ __align__(16) float sa[2][64 * LDT];
    __shared__ __align__(16) float sb[2][128 * LDT];
    const int tid = threadIdx.x, wave = tid >> 5, lane = tid & 31;
    const int lrow = lane & 15, kh = lane >> 4;
    const int gm0 = blockIdx.y * 64;
    const int gn0 = blockIdx.x * 128;
    const int m0 = (wave & 3) * 16;
    const int n0 = (wave >> 2) * 64;
    const int srow = tid >> 3;        // staging row base (adds 32 per pass)
    const int sq = (tid & 7) * 4;     // staging K quad

    auto stage = [&](int k0, int buf) {
#pragma unroll
        for (int r = 0; r < 2; r++) {
            const int row = srow + r * 32;
            cp_g2l_16(X + (size_t)(gm0 + row) * DIM + k0 + sq, &sa[buf][row * LDT + sq]);
        }
#pragma unroll
        for (int r = 0; r < 4; r++) {
            const int row = srow + r * 32;
            cp_g2l_16(W + (size_t)(gn0 + row) * DIM + k0 + sq, &sb[buf][row * LDT + sq]);
        }
    };

    v8f acc[4] = {};
    stage(0, 0);
    for (int k0 = 0; k0 < DIM; k0 += 32) {
        const int cur = (k0 >> 5) & 1;
        if (k0 + 32 < DIM) stage(k0 + 32, cur ^ 1);
#if USE_ASYNC
        if (k0 + 32 < DIM) __builtin_amdgcn_s_wait_asynccnt(6);  // retire current buffer's 6 copies
        else               __builtin_amdgcn_s_wait_asynccnt(0);
#endif
        __syncthreads();
#pragma unroll
        for (int kk = 0; kk < 32; kk += 4) {
            const v2f a = *(const v2f*)(&sa[cur][(m0 + lrow) * LDT + kk + kh * 2]);
#pragma unroll
            for (int c = 0; c < 4; c++) {
                const v2f b = *(const v2f*)(&sb[cur][(n0 + c * 16 + lrow) * LDT + kk + kh * 2]);
                acc[c] = wmma4(a, b, acc[c]);
            }
        }
        __syncthreads();
    }
#pragma unroll
    for (int c = 0; c < 4; c++) {
#pragma unroll
        for (int v = 0; v < 8; v++) {
            const int row = gm0 + m0 + v + kh * 8;
            const int col = gn0 + n0 + c * 16 + lrow;
            const size_t off = (size_t)row * DIM + col;
            if constexpr (FUSE) O[off] = O[off] + gate[row] * acc[c][v];
            else                O[off] = acc[c][v];
        }
    }
}

// ---------------- K3: excitation = mean(|slow|, axis=1) ----------------
__global__ __launch_bounds__(256) void excitation_kernel(const float* __restrict__ S, float* __restrict__ ex) {
    const int tid = threadIdx.x, wave = tid >> 5, lane = tid & 31;
    const int row = blockIdx.x * 8 + wave;
    const float* sr = S + (size_t)row * DIM;
    float s = 0.0f;
#pragma unroll
    for (int j = 0; j < 8; j++) {
        float4 v = *(const float4*)(sr + j * 128 + lane * 4);
        s += fabsf(v.x) + fabsf(v.y) + fabsf(v.z) + fabsf(v.w);
    }
    s = wave_sum(s);
    if (lane == 0) ex[row] = s * (1.0f / DIM);
}

// ---------------- K4: regulator MLP per row ----------------
__global__ __launch_bounds__(256) void regulator_kernel(
    const float* __restrict__ stress, const float* __restrict__ excit,
    const float* __restrict__ scal, const float* __restrict__ task_loss,
    const float* __restrict__ w1, const float* __restrict__ b1,
    const float* __restrict__ g,  const float* __restrict__ bn,
    const float* __restrict__ w2, const float* __restrict__ b2,
    float* __restrict__ met, float* __restrict__ sens, float* __restrict__ gate,
    float* __restrict__ met_sum) {
    const int i = blockIdx.x * 256 + threadIdx.x;
    const int lane = threadIdx.x & 31;
    const float f0 = stress[i];
    const float f1 = excit[i];
    const float f2 = sqrtf(scal[0]);
    const float f3 = task_loss[0];
    float z[16];
    float m = 0.0f;
#pragma unroll
    for (int j = 0; j < 16; j++) {
        z[j] = w1[j * 4 + 0] * f0 + w1[j * 4 + 1] * f1 + w1[j * 4 + 2] * f2 + w1[j * 4 + 3] * f3 + b1[j];
        m += z[j];
    }
    m *= (1.0f / 16.0f);
    float var = 0.0f;
#pragma unroll
    for (int j = 0; j < 16; j++) { float d = z[j] - m; var += d * d; }
    var *= (1.0f / 16.0f);
    const float inv = rsqrtf(var + 1e-5f);
    float h[16];
#pragma unroll
    for (int j = 0; j < 16; j++) h[j] = tanhf((z[j] - m) * inv * g[j] + bn[j]);
    float o[3];
#pragma unroll
    for (int c = 0; c < 3; c++) {
        float acc = b2[c];
#pragma unroll
        for (int j = 0; j < 16; j++) acc += w2[c * 16 + j] * h[j];
        o[c] = sigmoidf(acc);
    }
    met[i] = o[0];
    sens[i] = o[1];
    gate[i] = o[2];
    float ms = wave_sum(o[0]);
    if (lane == 0) atomicAdd(met_sum, ms);
}

// ---------------- K5: hebb partials = slow^T @ x (WMMA f32, K-sliced, async transpose staging) ----
// Block tile 64(M=slow col) x 64(N=x col), K-chunk 8192 per grid.z slice.
__global__ __launch_bounds__(256) void hebb_gemm_kernel(const float* __restrict__ S,
                                                        const float* __restrict__ X,
                                                        float* __restrict__ P) {
    __shared__ __align__(16) float sa[2][64 * LDT2];
    __shared__ __align__(16) float sb[2][64 * LDT2];
    const int tid = threadIdx.x, wave = tid >> 5, lane = tid & 31;
    const int lrow = lane & 15, kh = lane >> 4;
    const int gn0 = blockIdx.x * 64;   // n: columns of x
    const int gm0 = blockIdx.y * 64;   // m: columns of slow
    const int kbeg = blockIdx.z * (B_ROWS / 4);
    const int m0 = (wave >> 1) * 16;
    const int n0 = (wave & 1) * 32;
    const int scol = tid & 63;         // staging column (coalesced across lanes)
    const int skr0 = tid >> 6;         // staging k row base (adds 4 per pass)

    auto stage = [&](int k0, int buf) {
#pragma unroll
        for (int r = 0; r < 8; r++) {
            const int kr = skr0 + r * 4;
            // transpose in the copy: per-lane LDS destination [col][k]
            cp_g2l_4(S + (size_t)(k0 + kr) * DIM + gm0 + scol, &sa[buf][scol * LDT2 + kr]);
            cp_g2l_4(X + (size_t)(k0 + kr) * DIM + gn0 + scol, &sb[buf][scol * LDT2 + kr]);
        }
    };

    v8f acc[2] = {};
    stage(kbeg, 0);
    for (int it = 0; it < (B_ROWS / 4) / 32; it++) {
        const int k0 = kbeg + it * 32;
        const int cur = it & 1;
        const bool more = (it + 1) < (B_ROWS / 4) / 32;
        if (more) stage(k0 + 32, cur ^ 1);
#if USE_ASYNC
        if (more) __builtin_amdgcn_s_wait_asynccnt(16);  // retire current buffer's 16 copies
        else      __builtin_amdgcn_s_wait_asynccnt(0);
#endif
        __syncthreads();
#pragma unroll
        for (int kk = 0; kk < 32; kk += 4) {
            const v2f a = *(const v2f*)(&sa[cur][(m0 + lrow) * LDT2 + kk + kh * 2]);
#pragma unroll
            for (int c = 0; c < 2; c++) {
                const v2f b = *(const v2f*)(&sb[cur][(n0 + c * 16 + lrow) * LDT2 + kk + kh * 2]);
                acc[c] = wmma4(a, b, acc[c]);
            }
        }
        __syncthreads();
    }
    float* Pp = P + (size_t)blockIdx.z * DIM * DIM;
#pragma unroll
    for (int c = 0; c < 2; c++) {
#pragma unroll
        for (int v = 0; v < 8; v++) {
            const int row = gm0 + m0 + v + kh * 8;
            const int col = gn0 + n0 + c * 16 + lrow;
            Pp[(size_t)row * DIM + col] = acc[c][v];
        }
    }
}

// ---------------- K5b: W_fast_new = (W_fast + rate*tanh(hebb)) * 0.9995 ----------------
__global__ __launch_bounds__(256) void wfast_update_kernel(const float* __restrict__ P,
                                                           const float* __restrict__ Wf,
                                                           const float* __restrict__ scal,
                                                           float* __restrict__ outW) {
    const int i = blockIdx.x * 256 + threadIdx.x;
    const float hebb = (P[i] + P[i + DIM * DIM] + P[i + 2 * DIM * DIM] + P[i + 3 * DIM * DIM]) * (1.0f / B_ROWS);
    const float rate = 0.1f * (scal[1] * (1.0f / B_ROWS));
    outW[i] = (Wf[i] + rate * tanhf(hebb)) * 0.9995f;
}

// ---------------- K7: out = combined*sigmoid(beta*combined); y = layernorm(out) ----------------
__global__ __launch_bounds__(256) void silu_ln_kernel(const float* __restrict__ C,
                                                      const float* __restrict__ sens,
                                                      const float* __restrict__ lng,
                                                      const float* __restrict__ lnb,
                                                      float* __restrict__ Y) {
    const int tid = threadIdx.x, wave = tid >> 5, lane = tid & 31;
    const int row = blockIdx.x * 8 + wave;
    const float* cr = C + (size_t)row * DIM;
    const float beta = 0.5f + 2.0f * sens[row];
    float4 o[8];
    float s = 0.0f, s2 = 0.0f;
#pragma unroll
    for (int j = 0; j < 8; j++) {
        float4 v = *(const float4*)(cr + j * 128 + lane * 4);
        float4 t;
        t.x = v.x * sigmoidf(beta * v.x);
        t.y = v.y * sigmoidf(beta * v.y);
        t.z = v.z * sigmoidf(beta * v.z);
        t.w = v.w * sigmoidf(beta * v.w);
        o[j] = t;
        s  += t.x + t.y + t.z + t.w;
        s2 += t.x * t.x + t.y * t.y + t.z * t.z + t.w * t.w;
    }
    s = wave_sum(s);
    s2 = wave_sum(s2);
    const float m = s * (1.0f / DIM);
    const float var = s2 * (1.0f / DIM) - m * m;
    const float inv = rsqrtf(var + 1e-5f);
    float* yr = Y + (size_t)row * DIM;
#pragma unroll
    for (int j = 0; j < 8; j++) {
        const int col = j * 128 + lane * 4;
        float4 gv = *(const float4*)(lng + col);
        float4 bv = *(const float4*)(lnb + col);
        float4 r;
        r.x = (o[j].x - m) * inv * gv.x + bv.x;
        r.y = (o[j].y - m) * inv * gv.y + bv.y;
        r.z = (o[j].z - m) * inv * gv.z + bv.z;
        r.w = (o[j].w - m) * inv * gv.w + bv.w;
        *(float4*)(yr + col) = r;
    }
}

extern "C" void kernel_launch(void* const* d_in, const int* in_sizes, int n_in,
                              void* d_out, int out_size, void* d_ws, size_t ws_size,
                              hipStream_t stream) {
    const float* x         = (const float*)d_in[0];
    const float* task_loss = (const float*)d_in[1];
    const float* W_slow    = (const float*)d_in[2];
    const float* W_fast    = (const float*)d_in[3];
    const float* reg_w1    = (const float*)d_in[4];
    const float* reg_b1    = (const float*)d_in[5];
    const float* reg_g     = (const float*)d_in[6];
    const float* reg_bn    = (const float*)d_in[7];
    const float* reg_w2    = (const float*)d_in[8];
    const float* reg_b2    = (const float*)d_in[9];
    const float* ln_g      = (const float*)d_in[10];
    const float* ln_b      = (const float*)d_in[11];

    float* out   = (float*)d_out;
    float* y     = out;                                    // 32768*1024
    float* met   = out + (size_t)B_ROWS * DIM;             // 32768
    float* sens  = met + B_ROWS;                           // 32768
    float* gate  = sens + B_ROWS;                          // 32768
    float* wfast = gate + B_ROWS;                          // 1024*1024

    float* ws      = (float*)d_ws;
    float* slowbuf = ws;                                   // 32768*1024 (slow, later combined in-place)
    float* stress  = ws + (size_t)B_ROWS * DIM;            // 32768
    float* excit   = stress + B_ROWS;                      // 32768
    float* scal    = excit + B_ROWS;                       // [0]=||W||^2, [1]=sum(metabolism)
    float* hebbp   = scal + 8;                             // 4 * 1024*1024 partials

    (void)hipMemsetAsync(scal, 0, 2 * sizeof(float), stream);

    wnorm_kernel<<<256, 256, 0, stream>>>(W_slow, scal);
    rowstats_kernel<<<B_ROWS / 8, 256, 0, stream>>>(x, stress);
    gemm_xw_kernel<false><<<dim3(DIM / 128, B_ROWS / 64), 256, 0, stream>>>(x, W_slow, nullptr, slowbuf);
    excitation_kernel<<<B_ROWS / 8, 256, 0, stream>>>(slowbuf, excit);
    regulator_kernel<<<B_ROWS / 256, 256, 0, stream>>>(stress, excit, scal, task_loss,
                                                       reg_w1, reg_b1, reg_g, reg_bn,
                                                       reg_w2, reg_b2, met, sens, gate, &scal[1]);
    hebb_gemm_kernel<<<dim3(DIM / 64, DIM / 64, 4), 256, 0, stream>>>(slowbuf, x, hebbp);
    wfast_update_kernel<<<DIM * DIM / 256, 256, 0, stream>>>(hebbp, W_fast, scal, wfast);
    gemm_xw_kernel<true><<<dim3(DIM / 128, B_ROWS / 64), 256, 0, stream>>>(x, wfast, gate, slowbuf);
    silu_ln_kernel<<<B_ROWS / 8, 256, 0, stream>>>(slowbuf, sens, ln_g, ln_b, y);
}